// GCN_33088428048937
// MI455X (gfx1250) — compile-verified
//
#include <hip/hip_runtime.h>
#include <math.h>

typedef __attribute__((ext_vector_type(2))) float v2f;
typedef __attribute__((ext_vector_type(8))) float v8f;

#define N_NODES 100000
#define N_EDGES 1600000
#define IN_DIM 256
#define HID 128
#define N_GRAPHS 512

// ---------------- utility ----------------
__global__ void fill_kernel(float* __restrict__ p, float v, int n) {
    int i = blockIdx.x * blockDim.x + threadIdx.x;
    if (i < n) p[i] = v;
}

// deg[dst]++ per edge (deg pre-initialized to 1.0 for the self loop)
__global__ void deg_kernel(const int* __restrict__ dst, float* __restrict__ deg, int nE) {
    int e = blockIdx.x * blockDim.x + threadIdx.x;
    if (e < nE) atomicAdd(&deg[dst[e]], 1.0f);
}

__global__ void rsqrt_kernel(float* __restrict__ d, int n) {
    int i = blockIdx.x * blockDim.x + threadIdx.x;
    if (i < n) d[i] = __frsqrt_rn(d[i]);   // deg >= 1 always (self loop)
}

// ---------------- xw = x @ W1^T via V_WMMA_F32_16X16X4_F32 ----------------
// One wave per 16-row tile of x. K=256 stepped by 4; 8 accumulators cover N=128.
// A frag (16x4 f32): lane L holds A(M=L&15, K=kk+2*(L>>4)+{0,1}) in {a.x,a.y}
// B frag (4x16 f32): lane L holds B(K=kk+2*(L>>4)+{0,1}, N=nt*16+(L&15))
// C/D (16x16 f32):   reg r, lane L -> (M = r + 8*(L>>4), N = L&15)
__global__ __launch_bounds__(32) void gemm_xw_wmma(const float* __restrict__ x,
                                                   const float* __restrict__ W1,
                                                   float* __restrict__ xw) {
    const int tile = blockIdx.x;            // 0..6249  (100000/16)
    const int lane = threadIdx.x;           // 0..31
    const int mrow = tile * 16 + (lane & 15);
    const int kh   = (lane >> 4) * 2;       // 0 or 2

    v8f acc[8];
#pragma unroll
    for (int nt = 0; nt < 8; ++nt) acc[nt] = (v8f){0.f,0.f,0.f,0.f,0.f,0.f,0.f,0.f};

    const float* xrow = x + (size_t)mrow * IN_DIM;
    const int ncol = lane & 15;

    for (int kk = 0; kk < IN_DIM; kk += 4) {
        v2f a;
        a.x = xrow[kk + kh];
        a.y = xrow[kk + kh + 1];
#pragma unroll
        for (int nt = 0; nt < 8; ++nt) {
            const float* wrow = W1 + (size_t)(nt * 16 + ncol) * IN_DIM; // W1^T(k,n)=W1[n][k]
            v2f b;
            b.x = wrow[kk + kh];
            b.y = wrow[kk + kh + 1];
            acc[nt] = __builtin_amdgcn_wmma_f32_16x16x4_f32(
                false, a, false, b, (short)0, acc[nt], false, false);
        }
    }

    const int rbase = tile * 16 + (lane >> 4) * 8;   // M offset for this half-wave
#pragma unroll
    for (int nt = 0; nt < 8; ++nt) {
        const int col = nt * 16 + ncol;
#pragma unroll
        for (int r = 0; r < 8; ++r)
            xw[(size_t)(rbase + r) * HID + col] = acc[nt][r];
    }
}

// ---------------- layer-1 edge aggregation: h[dst] += norm * xw[src] ----------------
// one wave per edge, 4 features per lane (L2-resident scatter: xw/h fit in 192MB L2)
__global__ void agg1_kernel(const float* __restrict__ xw,
                            const int* __restrict__ src, const int* __restrict__ dst,
                            const float* __restrict__ dinv, float* __restrict__ h, int nE) {
    int t = blockIdx.x * blockDim.x + threadIdx.x;
    int e = t >> 5, lane = t & 31;
    if (e >= nE) return;
    int s = src[e], d = dst[e];
    float norm = dinv[s] * dinv[d];
    float4 v = ((const float4*)(xw + (size_t)s * HID))[lane];
    float* hd = h + (size_t)d * HID + lane * 4;
    atomicAdd(hd + 0, norm * v.x);
    atomicAdd(hd + 1, norm * v.y);
    atomicAdd(hd + 2, norm * v.z);
    atomicAdd(hd + 3, norm * v.w);
}

// self-loop contribution + tanh, fused elementwise over N_NODES*HID
__global__ void selfloop_tanh_kernel(const float* __restrict__ xw,
                                     const float* __restrict__ dinv,
                                     float* __restrict__ h, int n) {
    int i = blockIdx.x * blockDim.x + threadIdx.x;
    if (i >= n) return;
    float di = dinv[i >> 7];              // HID == 128
    h[i] = tanhf(h[i] + di * di * xw[i]);
}

// ---------------- layer 2: xw2[i] = dot(h[i,:], W2) ----------------
__global__ void dot_w2_kernel(const float* __restrict__ h, const float* __restrict__ W2,
                              float* __restrict__ xw2, int nNodes) {
    int t = blockIdx.x * blockDim.x + threadIdx.x;
    int node = t >> 5, lane = t & 31;
    if (node >= nNodes) return;
    float4 a = ((const float4*)(h + (size_t)node * HID))[lane];
    float4 b = ((const float4*)W2)[lane];
    float v = a.x * b.x + a.y * b.y + a.z * b.z + a.w * b.w;
#pragma unroll
    for (int off = 16; off > 0; off >>= 1)
        v += __shfl_down(v, off, 32);
    if (lane == 0) xw2[node] = v;
}

__global__ void agg2_kernel(const float* __restrict__ xw2,
                            const int* __restrict__ src, const int* __restrict__ dst,
                            const float* __restrict__ dinv, float* __restrict__ sacc, int nE) {
    int e = blockIdx.x * blockDim.x + threadIdx.x;
    if (e >= nE) return;
    int s = src[e], d = dst[e];
    atomicAdd(&sacc[d], dinv[s] * dinv[d] * xw2[s]);
}

// self loop + /HID + mean-pool accumulation
__global__ void pool_kernel(const float* __restrict__ sacc, const float* __restrict__ xw2,
                            const float* __restrict__ dinv, const int* __restrict__ batch,
                            float* __restrict__ sums, float* __restrict__ cnt, int nNodes) {
    int i = blockIdx.x * blockDim.x + threadIdx.x;
    if (i >= nNodes) return;
    float di = dinv[i];
    float s = (sacc[i] + di * di * xw2[i]) * (1.0f / (float)HID);
    int b = batch[i];
    atomicAdd(&sums[b], s);
    atomicAdd(&cnt[b], 1.0f);
}

// pooled -> softplus(-pooled*label) -> mean, single block of 512
__global__ void finalize_kernel(const float* __restrict__ sums, const float* __restrict__ cnt,
                                const float* __restrict__ label, float* __restrict__ out) {
    __shared__ float red[N_GRAPHS];
    int i = threadIdx.x;
    float c = cnt[i];
    float pooled = (c > 0.0f) ? sums[i] / fmaxf(c, 1.0f) : 0.0f;
    float z = -pooled * label[i];
    float sp = fmaxf(z, 0.0f) + log1pf(expf(-fabsf(z)));   // stable softplus
    red[i] = sp;
    __syncthreads();
#pragma unroll
    for (int stride = N_GRAPHS / 2; stride > 0; stride >>= 1) {
        if (i < stride) red[i] += red[i + stride];
        __syncthreads();
    }
    if (i == 0) out[0] = red[0] / (float)N_GRAPHS;
}

// ---------------- launch ----------------
extern "C" void kernel_launch(void* const* d_in, const int* in_sizes, int n_in,
                              void* d_out, int out_size, void* d_ws, size_t ws_size,
                              hipStream_t stream) {
    const float* x     = (const float*)d_in[0];
    const int*   ei    = (const int*)d_in[1];
    const int*   batch = (const int*)d_in[2];
    const float* label = (const float*)d_in[3];
    const float* W1    = (const float*)d_in[4];
    const float* W2    = (const float*)d_in[5];
    float* out = (float*)d_out;

    const int* src = ei;
    const int* dst = ei + N_EDGES;

    float* ws = (float*)d_ws;
    size_t off = 0;
    float* dinv = ws + off; off += 100096;                 // deg -> dinv in place
    float* xw   = ws + off; off += (size_t)N_NODES * HID;  // 12.8M floats
    float* h    = ws + off; off += (size_t)N_NODES * HID;  // 12.8M floats
    float* xw2  = ws + off; off += 100096;
    float* sacc = ws + off; off += 100096;
    float* sums = ws + off; off += N_GRAPHS;
    float* cnt  = ws + off; off += N_GRAPHS;

    const int TB = 256;
    int gN   = (N_NODES + TB - 1) / TB;
    int gE   = (N_EDGES + TB - 1) / TB;
    int gNH  = (N_NODES * HID + TB - 1) / TB;       // 50000
    int gE32 = (int)(((size_t)N_EDGES * 32 + TB - 1) / TB);   // 200000
    int gN32 = (int)(((size_t)N_NODES * 32 + TB - 1) / TB);   // 12500

    // init accumulators (ws/d_out are poisoned; zero everything we accumulate into)
    fill_kernel<<<gN, TB, 0, stream>>>(dinv, 1.0f, N_NODES);             // self-loop degree
    fill_kernel<<<gNH, TB, 0, stream>>>(h, 0.0f, N_NODES * HID);
    fill_kernel<<<gN, TB, 0, stream>>>(sacc, 0.0f, N_NODES);
    fill_kernel<<<(2 * N_GRAPHS + TB - 1) / TB, TB, 0, stream>>>(sums, 0.0f, 2 * N_GRAPHS);

    // degrees -> dinv
    deg_kernel<<<gE, TB, 0, stream>>>(dst, dinv, N_EDGES);
    rsqrt_kernel<<<gN, TB, 0, stream>>>(dinv, N_NODES);

    // layer 1
    gemm_xw_wmma<<<N_NODES / 16, 32, 0, stream>>>(x, W1, xw);
    agg1_kernel<<<gE32, TB, 0, stream>>>(xw, src, dst, dinv, h, N_EDGES);
    selfloop_tanh_kernel<<<gNH, TB, 0, stream>>>(xw, dinv, h, N_NODES * HID);

    // layer 2 + pooling + loss
    dot_w2_kernel<<<gN32, TB, 0, stream>>>(h, W2, xw2, N_NODES);
    agg2_kernel<<<gE, TB, 0, stream>>>(xw2, src, dst, dinv, sacc, N_EDGES);
    pool_kernel<<<gN, TB, 0, stream>>>(sacc, xw2, dinv, batch, sums, cnt, N_NODES);
    finalize_kernel<<<1, N_GRAPHS, 0, stream>>>(sums, cnt, label, out);
}